// SinkhornDistance_17540646437408
// MI455X (gfx1250) — compile-verified
//
#include <hip/hip_runtime.h>

// CDNA5 / gfx1250 — fp32 WMMA pairwise-sqdist + row-min + mean.
// C[b,i,j] = max(||x_i||^2 + ||y_j||^2 - 2 x_i.y_j, 0); out[b] = mean_i min_j C.

typedef __attribute__((ext_vector_type(2))) float v2f;
typedef __attribute__((ext_vector_type(8))) float v8f;

#define B_DIM   4
#define N_DIM   4096
#define M_DIM   4096
#define D_DIM   256
#define KSTEPS  (D_DIM / 4)        // 64 WMMA k-steps of K=4
#define TILE_J  16
#define NJT     (M_DIM / TILE_J)   // 256 target tiles
#define WAVES   8
#define ROWS_PER_BLOCK (WAVES * 16) // 128
#define NTILES_I (N_DIM / ROWS_PER_BLOCK) // 32
#define LDS_STRIDE 260             // 256 + 4 pad floats -> conflict-free B-frag ds_load_b64

__global__ __launch_bounds__(256, 1)
void ot_mincost_wmma(const float* __restrict__ src,
                     const float* __restrict__ tgt,
                     float* __restrict__ partial /* [B][NTILES_I] */) {
    __shared__ float tbuf[2][TILE_J * LDS_STRIDE];
    __shared__ float ynorm[2][TILE_J];
    __shared__ float wsum[WAVES];

    const int tid  = threadIdx.x;
    const int wave = tid >> 5;
    const int lane = tid & 31;
    const int l16  = lane & 15;
    const int half = lane >> 4;            // 0: K={0,1}, 1: K={2,3} per §7.12.2

    const int b       = blockIdx.y;
    const int rowBase = blockIdx.x * ROWS_PER_BLOCK + wave * 16;
    const int row     = rowBase + l16;

    const float* __restrict__ srcRow = src + ((size_t)b * N_DIM + row) * D_DIM;
    const float* __restrict__ tgtB   = tgt + (size_t)b * M_DIM * D_DIM;

    // ---- A fragments: resident in registers for the whole kernel (128 VGPRs) ----
    v2f areg[KSTEPS];
#pragma unroll
    for (int k = 0; k < KSTEPS; ++k)
        areg[k] = *(const v2f*)(srcRow + 4 * k + 2 * half);

    // ||x_row||^2 from registers: each lane has half the row; pair with lane^16.
    float xs = 0.f;
#pragma unroll
    for (int k = 0; k < KSTEPS; ++k)
        xs += areg[k].x * areg[k].x + areg[k].y * areg[k].y;
    xs += __shfl_xor(xs, 16, 32);          // xs = ||src[rowBase + l16]||^2

    // Accumulator-row v of the C tile maps to source row (v + 8*half).
    float xnv[8];
#pragma unroll
    for (int v = 0; v < 8; ++v)
        xnv[v] = __shfl(xs, v + 8 * half, 32);

    // ---- cooperative target-tile loader mapping: thread t -> row t/16, 16 floats ----
    const int ldRow = tid >> 4;            // 0..15 (target tile row = column j index)
    const int ldCol = (tid & 15) * 16;     // 0..240

    float rowmin[8];
#pragma unroll
    for (int v = 0; v < 8; ++v) rowmin[v] = 3.402823466e38f;

    float4 st0, st1, st2, st3;

    // Prologue: stage tile 0 into buffer 0.
    {
        const float* p = tgtB + (size_t)ldRow * D_DIM + ldCol;
        st0 = *(const float4*)(p + 0);
        st1 = *(const float4*)(p + 4);
        st2 = *(const float4*)(p + 8);
        st3 = *(const float4*)(p + 12);
        float* q = &tbuf[0][ldRow * LDS_STRIDE + ldCol];
        *(float4*)(q + 0)  = st0;
        *(float4*)(q + 4)  = st1;
        *(float4*)(q + 8)  = st2;
        *(float4*)(q + 12) = st3;
        float ys = st0.x*st0.x + st0.y*st0.y + st0.z*st0.z + st0.w*st0.w
                 + st1.x*st1.x + st1.y*st1.y + st1.z*st1.z + st1.w*st1.w
                 + st2.x*st2.x + st2.y*st2.y + st2.z*st2.z + st2.w*st2.w
                 + st3.x*st3.x + st3.y*st3.y + st3.z*st3.z + st3.w*st3.w;
        ys += __shfl_xor(ys, 1, 32);
        ys += __shfl_xor(ys, 2, 32);
        ys += __shfl_xor(ys, 4, 32);
        ys += __shfl_xor(ys, 8, 32);
        if (l16 == 0) ynorm[0][ldRow] = ys;
    }
    __syncthreads();

    int cur = 0;
    for (int jt = 0; jt < NJT; ++jt) {
        // Issue global loads for tile jt+1 now; they complete under the WMMA chain.
        if (jt + 1 < NJT) {
            const float* p = tgtB + ((size_t)(jt + 1) * TILE_J + ldRow) * D_DIM + ldCol;
            st0 = *(const float4*)(p + 0);
            st1 = *(const float4*)(p + 4);
            st2 = *(const float4*)(p + 8);
            st3 = *(const float4*)(p + 12);
        }
        if (jt + 2 < NJT) {
            __builtin_prefetch(tgtB + ((size_t)(jt + 2) * TILE_J + ldRow) * D_DIM + ldCol, 0, 0);
        }

        const float yn = ynorm[cur][l16];
        const float* bbase = &tbuf[cur][l16 * LDS_STRIDE + 2 * half];

        // 16x16 xy tile: 64 V_WMMA_F32_16X16X4_F32 in two independent accumulation
        // chains (even/odd k), with B fragments software-pipelined one pair ahead so
        // DS loads stay >=1 deep in flight under the WMMA issue stream.
        v8f acc0 = {0.f, 0.f, 0.f, 0.f, 0.f, 0.f, 0.f, 0.f};
        v8f acc1 = {0.f, 0.f, 0.f, 0.f, 0.f, 0.f, 0.f, 0.f};
        v2f b0 = *(const v2f*)(bbase + 0);
        v2f b1 = *(const v2f*)(bbase + 4);
#pragma unroll
        for (int k = 0; k < KSTEPS; k += 2) {
            const int kn = (k + 2 < KSTEPS) ? (k + 2) : k;   // clamp: stays in-bounds
            v2f n0 = *(const v2f*)(bbase + 4 * kn);
            v2f n1 = *(const v2f*)(bbase + 4 * (kn + 1));
            acc0 = __builtin_amdgcn_wmma_f32_16x16x4_f32(
                false, areg[k], false, b0, (short)0, acc0, false, false);
            acc1 = __builtin_amdgcn_wmma_f32_16x16x4_f32(
                false, areg[k + 1], false, b1, (short)0, acc1, false, false);
            b0 = n0;
            b1 = n1;
        }

        // Fused epilogue: clamped squared distance, running row-min.
#pragma unroll
        for (int v = 0; v < 8; ++v) {
            float xy = acc0[v] + acc1[v];
            float c = fmaxf(xnv[v] + yn - 2.0f * xy, 0.0f);
            rowmin[v] = fminf(rowmin[v], c);
        }

        // Publish next tile into the other buffer; single barrier per tile.
        if (jt + 1 < NJT) {
            float* q = &tbuf[cur ^ 1][ldRow * LDS_STRIDE + ldCol];
            *(float4*)(q + 0)  = st0;
            *(float4*)(q + 4)  = st1;
            *(float4*)(q + 8)  = st2;
            *(float4*)(q + 12) = st3;
            float ys = st0.x*st0.x + st0.y*st0.y + st0.z*st0.z + st0.w*st0.w
                     + st1.x*st1.x + st1.y*st1.y + st1.z*st1.z + st1.w*st1.w
                     + st2.x*st2.x + st2.y*st2.y + st2.z*st2.z + st2.w*st2.w
                     + st3.x*st3.x + st3.y*st3.y + st3.z*st3.z + st3.w*st3.w;
            ys += __shfl_xor(ys, 1, 32);
            ys += __shfl_xor(ys, 2, 32);
            ys += __shfl_xor(ys, 4, 32);
            ys += __shfl_xor(ys, 8, 32);
            if (l16 == 0) ynorm[cur ^ 1][ldRow] = ys;
        }
        __syncthreads();
        cur ^= 1;
    }

    // Row-min finalization: min across the 16 lanes of each half (columns mod 16).
#pragma unroll
    for (int v = 0; v < 8; ++v) {
        rowmin[v] = fminf(rowmin[v], __shfl_xor(rowmin[v], 1, 32));
        rowmin[v] = fminf(rowmin[v], __shfl_xor(rowmin[v], 2, 32));
        rowmin[v] = fminf(rowmin[v], __shfl_xor(rowmin[v], 4, 32));
        rowmin[v] = fminf(rowmin[v], __shfl_xor(rowmin[v], 8, 32));
    }
    // Lanes 0-15 hold mins of rows v; lanes 16-31 of rows v+8. Sum 16 rows per wave.
    float s = 0.f;
#pragma unroll
    for (int v = 0; v < 8; ++v) s += rowmin[v];
    s += __shfl_xor(s, 16, 32);
    if (lane == 0) wsum[wave] = s;
    __syncthreads();
    if (tid == 0) {
        float t = 0.f;
#pragma unroll
        for (int w = 0; w < WAVES; ++w) t += wsum[w];
        partial[b * NTILES_I + blockIdx.x] = t;
    }
}

// Deterministic final reduction: out[b] = (sum of 32 partials) / N.
__global__ void ot_finalize(const float* __restrict__ partial,
                            float* __restrict__ out) {
    int b = threadIdx.x;
    if (b < B_DIM) {
        float s = 0.f;
        for (int t = 0; t < NTILES_I; ++t) s += partial[b * NTILES_I + t];
        out[b] = s * (1.0f / (float)N_DIM);
    }
}

extern "C" void kernel_launch(void* const* d_in, const int* in_sizes, int n_in,
                              void* d_out, int out_size, void* d_ws, size_t ws_size,
                              hipStream_t stream) {
    const float* src = (const float*)d_in[0];   // [B, N, D] fp32
    const float* tgt = (const float*)d_in[1];   // [B, M, D] fp32
    float* out = (float*)d_out;                 // [B] fp32
    float* partial = (float*)d_ws;              // B * NTILES_I = 128 floats

    dim3 grid(NTILES_I, B_DIM);                 // 32 x 4 blocks
    ot_mincost_wmma<<<grid, 256, 0, stream>>>(src, tgt, partial);
    ot_finalize<<<1, 32, 0, stream>>>(partial, out);
}